// LengthRegulator_88235808129082
// MI455X (gfx1250) — compile-verified
//
#include <hip/hip_runtime.h>
#include <hip/hip_bf16.h>

// Dims fixed by setup_inputs(): N=16, L=512, E=H=256, K=3, T=2048.
#define NB   16
#define LL   512
#define CH   256
#define KG   768          // 3*256 GEMM reduction dim
#define TT   2048
#define AS_STRIDE 772     // 768 + 4 pad (avoid 64-bank conflicts)
#define LN_STRIDE 260     // 256 + 4 pad

typedef float v2f __attribute__((ext_vector_type(2)));
typedef float v4f __attribute__((ext_vector_type(4)));
typedef float v8f __attribute__((ext_vector_type(8)));

// -------- weight repack: w[H,E,K] -> B'[c/4][h][c%4]  (c = k*256 + e) -------
// Destination-indexed so the 768KB of writes stay coalesced; each lane's two
// B-fragment floats for one WMMA chunk become adjacent (one global_load_b64).
__global__ void repackW(const float* __restrict__ w, float* __restrict__ Bm) {
    int i   = blockIdx.x * 256 + threadIdx.x;   // dest index over 192*1024
    int kq  = i >> 10;                          // k-chunk 0..191
    int rem = i & 1023;
    int h   = rem >> 2;                         // output channel
    int c   = (kq << 2) + (rem & 3);            // GEMM k index 0..767
    int ktap = c >> 8;                          // conv tap 0..2
    int e    = c & 255;                         // input channel
    Bm[i] = w[h * KG + e * 3 + ktap];
}

// -------- fused conv1d(K=3,pad=1) as WMMA GEMM + bias + LayerNorm + ReLU -----
// block = 512 threads (16 waves); each block -> 16 rows x 256 cols of output.
__launch_bounds__(512)
__global__ void conv_ln_relu(const float* __restrict__ in,   // [N*L, 256]
                             const float* __restrict__ Bm,   // [192][256][4]
                             const float* __restrict__ bias, // [256]
                             const float* __restrict__ g,    // [256]
                             const float* __restrict__ bb,   // [256]
                             float* __restrict__ out) {      // [N*L, 256]
    __shared__ float As[16 * AS_STRIDE];   // 49,408 B; reused for LN buffer

    const int tid = threadIdx.x;
    const int rt  = blockIdx.x;            // row-tile: rows rt*16 .. rt*16+15
    const int n   = rt >> 5;               // 32 tiles per batch entry
    const int l0  = (rt & 31) << 4;

    // Stage im2col A tile [16 rows x 768 cols] with zero padding at l edges.
    {
        const int r  = tid >> 5;           // row 0..15 (one wave per row)
        const int ln = tid & 31;
        for (int j = 0; j < 24; ++j) {
            int c    = ln + (j << 5);      // 0..767, coalesced
            int ktap = c >> 8;             // 0..2
            int e    = c & 255;
            int lsrc = l0 + r + ktap - 1;
            float v  = 0.f;
            if ((unsigned)lsrc < (unsigned)LL)
                v = in[(((n << 9) + lsrc) << 8) + e];
            As[r * AS_STRIDE + c] = v;
        }
    }
    __syncthreads();

    const int wave = tid >> 5;
    const int lane = tid & 31;
    const int base = (lane >> 4) << 1;     // 0 (lanes 0-15) or 2 (lanes 16-31)
    const int mrow = lane & 15;
    const int col  = (wave << 4) + mrow;   // output channel 0..255

    // B'-fragment pointer: floats (kk+base, kk+base+1) for channel `col`
    // live at Bm[kk*256 + col*4 + base] -> single 8-byte load per WMMA.
    const float* __restrict__ Bp = Bm + (col << 2) + base;
    const float* Ap = As + mrow * AS_STRIDE + base;

    v8f acc = {0.f, 0.f, 0.f, 0.f, 0.f, 0.f, 0.f, 0.f};
    for (int kk = 0; kk < KG; kk += 4) {
        v2f a, b;
        a.x = Ap[kk];
        a.y = Ap[kk + 1];
        b   = *(const v2f*)(Bp + (kk << 8));
        acc = __builtin_amdgcn_wmma_f32_16x16x4_f32(
                  false, a, false, b, (short)0, acc, false, false);
    }

    const float bv = bias[col];
    __syncthreads();                       // GEMM done everywhere; reuse LDS
    for (int r = 0; r < 8; ++r) {
        int row = r + ((lane >> 4) << 3);  // C/D layout: VGPR r -> M=r / r+8
        As[row * LN_STRIDE + col] = acc[r] + bv;
    }
    __syncthreads();

    // LayerNorm over 256 channels + ReLU; wave w handles tile row w.
    {
        const int row = wave;
        float vals[8];
        float s = 0.f, ss = 0.f;
        for (int j = 0; j < 8; ++j) {
            float v = As[row * LN_STRIDE + lane + (j << 5)];
            vals[j] = v; s += v; ss += v * v;
        }
        for (int off = 16; off; off >>= 1) {
            s  += __shfl_xor(s,  off, 32);
            ss += __shfl_xor(ss, off, 32);
        }
        float mu  = s * (1.0f / 256.0f);
        float var = ss * (1.0f / 256.0f) - mu * mu;
        float inv = rsqrtf(var + 1e-5f);
        int grow  = (rt << 4) + row;
        for (int j = 0; j < 8; ++j) {
            int c   = lane + (j << 5);
            float y = (vals[j] - mu) * inv * g[c] + bb[c];
            out[(grow << 8) + c] = fmaxf(y, 0.f);
        }
    }
}

// -------- h2 @ lin_w + lin_b, ReLU -> duration [N*L] ------------------------
__global__ void linear_relu(const float* __restrict__ h,
                            const float* __restrict__ w,   // [256]
                            const float* __restrict__ b,   // [1]
                            float* __restrict__ dur) {
    int row  = blockIdx.x * 8 + (threadIdx.x >> 5);
    int lane = threadIdx.x & 31;
    const float* hr = h + (row << 8);
    float s = 0.f;
    for (int j = 0; j < 8; ++j) {
        int c = lane + (j << 5);
        s += hr[c] * w[c];
    }
    for (int off = 16; off; off >>= 1) s += __shfl_xor(s, off, 32);
    if (lane == 0) dur[row] = fmaxf(s + b[0], 0.f);
}

// -------- length regulator: gather x rows by cumsum(target) -----------------
// grid (T/4, N), block 256. Scan target[n,:] in LDS, binary-search frame.
__global__ void regulate(const float* __restrict__ x,   // [N, L, 256]
                         const int* __restrict__ tgt,   // [N, L]
                         float* __restrict__ out) {     // [N, T, 256]
    __shared__ int cs[LL];
    const int n = blockIdx.y;
    const int* tn = tgt + (n << 9);
    cs[threadIdx.x]       = tn[threadIdx.x];
    cs[threadIdx.x + 256] = tn[threadIdx.x + 256];
    __syncthreads();
    // Hillis-Steele inclusive scan over 512 ints with 256 threads.
    for (int off = 1; off < LL; off <<= 1) {
        int i0 = threadIdx.x, i1 = threadIdx.x + 256;
        int v0 = (i0 >= off) ? cs[i0 - off] : 0;
        int v1 = (i1 >= off) ? cs[i1 - off] : 0;
        __syncthreads();
        cs[i0] += v0; cs[i1] += v1;
        __syncthreads();
    }
    const int total = cs[LL - 1];
    const int t = (blockIdx.x << 2) + (threadIdx.x >> 6);   // frame index
    const int q = threadIdx.x & 63;                          // float4 slot
    v4f v = {0.f, 0.f, 0.f, 0.f};
    if (t < total) {
        int lo = 0, hi = LL;
        while (lo < hi) {                  // first j with cs[j] > t
            int mid = (lo + hi) >> 1;
            if (cs[mid] <= t) lo = mid + 1; else hi = mid;
        }
        v = ((const v4f*)(x + (((n << 9) + lo) << 8)))[q];
    }
    // Write-once 32MB stream: nontemporal store keeps L2 for B/x reuse.
    __builtin_nontemporal_store(v, (v4f*)(out + (((n << 11) + t) << 8)) + q);
}

extern "C" void kernel_launch(void* const* d_in, const int* in_sizes, int n_in,
                              void* d_out, int out_size, void* d_ws, size_t ws_size,
                              hipStream_t stream) {
    (void)in_sizes; (void)n_in; (void)out_size; (void)d_ws; (void)ws_size;

    const float* x   = (const float*)d_in[0];
    const float* w1  = (const float*)d_in[1];
    const float* b1  = (const float*)d_in[2];
    const float* g1  = (const float*)d_in[3];
    const float* bb1 = (const float*)d_in[4];
    const float* w2  = (const float*)d_in[5];
    const float* b2  = (const float*)d_in[6];
    const float* g2  = (const float*)d_in[7];
    const float* bb2 = (const float*)d_in[8];
    const float* lw  = (const float*)d_in[9];
    const float* lb  = (const float*)d_in[10];
    const int*   tgt = (const int*)d_in[11];
    // d_in[12] = mel_max_len (2048, hardcoded)

    float* out = (float*)d_out;                 // [16,2048,256] = 8,388,608
    float* dur = out + 8388608;                 // [16,512]

    // Scratch carved from the front of the output region; all of it is
    // consumed before `regulate` overwrites the full output. No d_ws needed.
    float* h1  = out;                           // 2,097,152 floats
    float* h2  = out + 2097152;                 // 2,097,152 floats
    float* Bm1 = out + 4194304;                 //   196,608 floats
    float* Bm2 = out + 4390912;                 //   196,608 floats

    repackW<<<768, 256, 0, stream>>>(w1, Bm1);
    repackW<<<768, 256, 0, stream>>>(w2, Bm2);
    conv_ln_relu<<<512, 512, 0, stream>>>(x,  Bm1, b1, g1, bb1, h1);
    conv_ln_relu<<<512, 512, 0, stream>>>(h1, Bm2, b2, g2, bb2, h2);
    linear_relu<<<1024, 256, 0, stream>>>(h2, lw, lb, dur);
    regulate<<<dim3(TT / 4, NB), 256, 0, stream>>>(x, tgt, out);
}